// SimilarityContrastiveLoss_24008867185353
// MI455X (gfx1250) — compile-verified
//
#include <hip/hip_runtime.h>
#include <math.h>

typedef float v2f __attribute__((ext_vector_type(2)));
typedef float v8f __attribute__((ext_vector_type(8)));

#define BN   4096      // batch
#define DD   1024      // feature dim
#define EPSV 1e-8f
#define MARG 0.5f

__device__ __forceinline__ v2f ldf2(const float* __restrict__ p, int idx) {
    return *(const v2f*)(p + idx);
}

__device__ __forceinline__ v8f wmma_f32(v2f a, v2f b, v8f c) {
    return __builtin_amdgcn_wmma_f32_16x16x4_f32(false, a, false, b, (short)0, c, false, false);
}

// ---------------- Kernel 1: row norms of x1 and x2 -> ws[0..8191] ----------------
__global__ void __launch_bounds__(256) norms_kernel(const float* __restrict__ x1,
                                                    const float* __restrict__ x2,
                                                    float* __restrict__ ws) {
    __shared__ float red[256];
    const int row = blockIdx.x;  // 0..8191
    const float* src = (row < BN) ? (x1 + (size_t)row * DD)
                                  : (x2 + (size_t)(row - BN) * DD);
    float s = 0.f;
    #pragma unroll
    for (int t = threadIdx.x; t < DD; t += 256) {
        float v = src[t];
        s += v * v;
    }
    red[threadIdx.x] = s;
    __syncthreads();
    for (int off = 128; off > 0; off >>= 1) {
        if ((int)threadIdx.x < off) red[threadIdx.x] += red[threadIdx.x + off];
        __syncthreads();
    }
    if (threadIdx.x == 0) ws[row] = sqrtf(red[0]);
}

// ------- Kernel 2: WMMA fp32 GEMM (x1 @ x2^T) fused with contrastive loss -------
// Workgroup tile: 128 rows x 128 cols. 8 waves arranged 4x2; wave tile 32x64.
// kdim is passed at runtime so the K-loop stays rolled (I-cache resident body).
__global__ void __launch_bounds__(256) gemm_loss_kernel(const float* __restrict__ x1,
                                                        const float* __restrict__ x2,
                                                        const float* __restrict__ ws,
                                                        float* __restrict__ partials,
                                                        int kdim) {
    __shared__ float red[256];
    const int tid   = threadIdx.x;
    const int wid   = tid >> 5;      // wave id 0..7 (wave32)
    const int lane  = tid & 31;
    const int half  = lane >> 4;     // 0: lanes 0-15, 1: lanes 16-31
    const int mrow  = lane & 15;

    const int waveM = wid >> 1;      // 0..3 (32-row strips)
    const int waveN = wid & 1;       // 0..1 (64-col strips)

    const int rowW = blockIdx.x * 128 + waveM * 32;
    const int colW = blockIdx.y * 128 + waveN * 64;

    // A (16x4 f32): lane holds row M=mrow; VGPR0=K(2*half), VGPR1=K(2*half+1)
    // B (4x16 f32): lane holds col N=mrow; same K mapping (B[k][n] = x2[n][k])
    const int ia0 = (rowW +  0 + mrow) * DD + 2 * half;
    const int ia1 = ia0 + 16 * DD;
    const int ib0 = (colW +  0 + mrow) * DD + 2 * half;
    const int ib1 = ib0 + 16 * DD;
    const int ib2 = ib0 + 32 * DD;
    const int ib3 = ib0 + 48 * DD;

    v8f acc[2][4] = {};   // [mtile][ntile], zero-initialized

    // 1-deep software pipeline: next k-step's 6 fragments in flight while the
    // current 8 WMMAs execute.
    v2f A0 = ldf2(x1, ia0), A1 = ldf2(x1, ia1);
    v2f B0 = ldf2(x2, ib0), B1 = ldf2(x2, ib1);
    v2f B2 = ldf2(x2, ib2), B3 = ldf2(x2, ib3);

    #pragma unroll 4
    for (int kb = 4; kb < kdim; kb += 4) {
        v2f A0n = ldf2(x1, ia0 + kb);
        v2f A1n = ldf2(x1, ia1 + kb);
        v2f B0n = ldf2(x2, ib0 + kb);
        v2f B1n = ldf2(x2, ib1 + kb);
        v2f B2n = ldf2(x2, ib2 + kb);
        v2f B3n = ldf2(x2, ib3 + kb);

        acc[0][0] = wmma_f32(A0, B0, acc[0][0]);
        acc[0][1] = wmma_f32(A0, B1, acc[0][1]);
        acc[0][2] = wmma_f32(A0, B2, acc[0][2]);
        acc[0][3] = wmma_f32(A0, B3, acc[0][3]);
        acc[1][0] = wmma_f32(A1, B0, acc[1][0]);
        acc[1][1] = wmma_f32(A1, B1, acc[1][1]);
        acc[1][2] = wmma_f32(A1, B2, acc[1][2]);
        acc[1][3] = wmma_f32(A1, B3, acc[1][3]);

        A0 = A0n; A1 = A1n;
        B0 = B0n; B1 = B1n; B2 = B2n; B3 = B3n;
    }

    // drain: last k-step
    acc[0][0] = wmma_f32(A0, B0, acc[0][0]);
    acc[0][1] = wmma_f32(A0, B1, acc[0][1]);
    acc[0][2] = wmma_f32(A0, B2, acc[0][2]);
    acc[0][3] = wmma_f32(A0, B3, acc[0][3]);
    acc[1][0] = wmma_f32(A1, B0, acc[1][0]);
    acc[1][1] = wmma_f32(A1, B1, acc[1][1]);
    acc[1][2] = wmma_f32(A1, B2, acc[1][2]);
    acc[1][3] = wmma_f32(A1, B3, acc[1][3]);

    // Fused epilogue: cos = dot / max(n1*n2, eps);
    // diag: (1 - cos), off-diag: max(cos - margin, 0)
    const float* n1 = ws;
    const float* n2 = ws + BN;
    float lsum = 0.f;
    #pragma unroll
    for (int tm = 0; tm < 2; ++tm) {
        #pragma unroll
        for (int tn = 0; tn < 4; ++tn) {
            #pragma unroll
            for (int r = 0; r < 8; ++r) {
                // C/D layout: VGPR r -> M = 8*half + r; N = mrow
                const int i = rowW + tm * 16 + half * 8 + r;
                const int j = colW + tn * 16 + mrow;
                const float dot   = acc[tm][tn][r];
                const float denom = fmaxf(n1[i] * n2[j], EPSV);
                const float cosv  = dot / denom;
                lsum += (i == j) ? (1.0f - cosv) : fmaxf(cosv - MARG, 0.0f);
            }
        }
    }

    // deterministic block reduction
    red[tid] = lsum;
    __syncthreads();
    for (int off = 128; off > 0; off >>= 1) {
        if (tid < off) red[tid] += red[tid + off];
        __syncthreads();
    }
    if (tid == 0) partials[blockIdx.y * gridDim.x + blockIdx.x] = red[0];
}

// ---------------- Kernel 3: deterministic final reduction ----------------
__global__ void __launch_bounds__(256) finalize_kernel(const float* __restrict__ partials,
                                                       float* __restrict__ out) {
    __shared__ float red[256];
    float s = 0.f;
    for (int t = threadIdx.x; t < 1024; t += 256) s += partials[t];
    red[threadIdx.x] = s;
    __syncthreads();
    for (int off = 128; off > 0; off >>= 1) {
        if ((int)threadIdx.x < off) red[threadIdx.x] += red[threadIdx.x + off];
        __syncthreads();
    }
    if (threadIdx.x == 0) out[0] = red[0] / (float)BN;
}

extern "C" void kernel_launch(void* const* d_in, const int* in_sizes, int n_in,
                              void* d_out, int out_size, void* d_ws, size_t ws_size,
                              hipStream_t stream) {
    const float* x1 = (const float*)d_in[0];
    const float* x2 = (const float*)d_in[1];
    float* ws       = (float*)d_ws;           // [0,4096): n1, [4096,8192): n2
    float* partials = ws + 2 * BN;            // [8192, 8192+1024)
    float* out      = (float*)d_out;

    norms_kernel<<<2 * BN, 256, 0, stream>>>(x1, x2, ws);

    dim3 grid(BN / 128, BN / 128);            // 32 x 32 workgroups
    gemm_loss_kernel<<<grid, 256, 0, stream>>>(x1, x2, ws, partials, DD);

    finalize_kernel<<<1, 256, 0, stream>>>(partials, out);
}